// GraphSAGE_82721070121705
// MI455X (gfx1250) — compile-verified
//
#include <hip/hip_runtime.h>

#define NNODES 40000
#define NEDGES 640000
#define NF 128
#define NGRAPHS 64

typedef __attribute__((ext_vector_type(2))) float v2f;
typedef __attribute__((ext_vector_type(8))) float v8f;

__global__ void zero_f32(float* __restrict__ p, int n) {
    int i = blockIdx.x * blockDim.x + threadIdx.x;
    if (i < n) p[i] = 0.0f;
}

__global__ void degree_kernel(const int* __restrict__ dst, float* __restrict__ deg) {
    int e = blockIdx.x * blockDim.x + threadIdx.x;
    if (e < NEDGES) atomicAdd(&deg[dst[e]], 1.0f);
}

// One wave32 per edge: lane l moves features [4l, 4l+4) via float4 load + 4 atomic adds.
__global__ void scatter_kernel(const float* __restrict__ x, const int* __restrict__ src,
                               const int* __restrict__ dst, float* __restrict__ agg) {
    int gtid = blockIdx.x * blockDim.x + threadIdx.x;
    int e    = gtid >> 5;
    int lane = gtid & 31;
    if (e >= NEDGES) return;
    int s = src[e];
    int d = dst[e];
    const float4 v = *(const float4*)(x + (size_t)s * NF + lane * 4);
    float* base = agg + (size_t)d * NF + lane * 4;
    atomicAdd(base + 0, v.x);
    atomicAdd(base + 1, v.y);
    atomicAdd(base + 2, v.z);
    atomicAdd(base + 3, v.w);
}

// Fused SAGE layer GEMM: hout = relu((agg/deg) @ Wl + x @ Wr + b), plus pooled
// epilogue: pooled[batch[row]][poolOff + col] += hout. One block = 16 rows x 128
// cols; 8 wave32s, each computing one 16x16 f32 WMMA tile, K looped by 4.
__global__ void __launch_bounds__(256)
sage_gemm(const float* __restrict__ agg, const float* __restrict__ deg,
          const float* __restrict__ xin, const float* __restrict__ Wl,
          const float* __restrict__ Wr, const float* __restrict__ bias,
          const int* __restrict__ batch, float* __restrict__ hout,
          float* __restrict__ pooled, int poolOff) {
    __shared__ float sA[16][NF + 4];   // pad -> conflict-free column reads
    __shared__ float sX[16][NF + 4];

    const int r0  = blockIdx.x * 16;
    const int tid = threadIdx.x;

    // Stage 16x128 normalized-agg and x tiles into LDS (8 elems per thread).
    for (int i = tid; i < 16 * NF; i += 256) {
        int r = i >> 7, c = i & (NF - 1);
        int row = r0 + r;
        float rdeg = 1.0f / fmaxf(deg[row], 1.0f);
        sA[r][c] = agg[(size_t)row * NF + c] * rdeg;
        sX[r][c] = xin[(size_t)row * NF + c];
    }
    __syncthreads();

    const int wave = tid >> 5;
    const int lane = tid & 31;
    const int half = lane >> 4;        // 0: lanes 0-15, 1: lanes 16-31
    const int mn   = lane & 15;        // A row / B,D col within tile
    const int kk   = half * 2;         // K sub-offset per ISA A/B layout
    const int cb   = wave * 16;        // column base of this wave's tile

    v8f acc = {};
    #pragma unroll 4
    for (int k = 0; k < NF; k += 4) {
        v2f a, b;
        a.x = sA[mn][k + kk];
        a.y = sA[mn][k + kk + 1];
        b.x = Wl[(k + kk) * NF + cb + mn];
        b.y = Wl[(k + kk + 1) * NF + cb + mn];
        acc = __builtin_amdgcn_wmma_f32_16x16x4_f32(false, a, false, b,
                                                    (short)0, acc, false, false);
    }
    #pragma unroll 4
    for (int k = 0; k < NF; k += 4) {
        v2f a, b;
        a.x = sX[mn][k + kk];
        a.y = sX[mn][k + kk + 1];
        b.x = Wr[(k + kk) * NF + cb + mn];
        b.y = Wr[(k + kk + 1) * NF + cb + mn];
        acc = __builtin_amdgcn_wmma_f32_16x16x4_f32(false, a, false, b,
                                                    (short)0, acc, false, false);
    }

    const float bv = bias[cb + mn];
    #pragma unroll
    for (int r = 0; r < 8; ++r) {
        int grow = r0 + r + half * 8;          // D: VGPR r holds M=r (+8 for hi half)
        float val = acc[r] + bv;
        val = val > 0.0f ? val : 0.0f;         // ReLU
        if (hout) hout[(size_t)grow * NF + cb + mn] = val;
        atomicAdd(&pooled[batch[grow] * (3 * NF) + poolOff + cb + mn], val);
    }
}

// out = relu(pooled[64,384] @ Wlin[384,128] + blin). 4 blocks x 8 waves.
__global__ void __launch_bounds__(256)
final_gemm(const float* __restrict__ pooled, const float* __restrict__ Wlin,
           const float* __restrict__ blin, float* __restrict__ out) {
    const int r0   = blockIdx.x * 16;
    const int tid  = threadIdx.x;
    const int wave = tid >> 5;
    const int lane = tid & 31;
    const int half = lane >> 4;
    const int mn   = lane & 15;
    const int kk   = half * 2;
    const int cb   = wave * 16;

    v8f acc = {};
    for (int k = 0; k < 3 * NF; k += 4) {
        v2f a, b;
        a.x = pooled[(r0 + mn) * (3 * NF) + k + kk];
        a.y = pooled[(r0 + mn) * (3 * NF) + k + kk + 1];
        b.x = Wlin[(k + kk) * NF + cb + mn];
        b.y = Wlin[(k + kk + 1) * NF + cb + mn];
        acc = __builtin_amdgcn_wmma_f32_16x16x4_f32(false, a, false, b,
                                                    (short)0, acc, false, false);
    }
    const float bv = blin[cb + mn];
    #pragma unroll
    for (int r = 0; r < 8; ++r) {
        int grow = r0 + r + half * 8;
        float val = acc[r] + bv;
        out[grow * NF + cb + mn] = val > 0.0f ? val : 0.0f;
    }
}

extern "C" void kernel_launch(void* const* d_in, const int* in_sizes, int n_in,
                              void* d_out, int out_size, void* d_ws, size_t ws_size,
                              hipStream_t stream) {
    const float* x    = (const float*)d_in[0];
    const int*   edge = (const int*)d_in[1];
    const int*   src  = edge;               // edge_index[0]
    const int*   dst  = edge + NEDGES;      // edge_index[1]
    const int*   batch= (const int*)d_in[2];
    const float* W1l  = (const float*)d_in[3];
    const float* b1   = (const float*)d_in[4];
    const float* W1r  = (const float*)d_in[5];
    const float* W2l  = (const float*)d_in[6];
    const float* b2   = (const float*)d_in[7];
    const float* W2r  = (const float*)d_in[8];
    const float* W3l  = (const float*)d_in[9];
    const float* b3   = (const float*)d_in[10];
    const float* W3r  = (const float*)d_in[11];
    const float* Wlin = (const float*)d_in[12];
    const float* blin = (const float*)d_in[13];
    float* out = (float*)d_out;

    // Workspace layout (floats): deg | agg | h1 | h2 | pooled  (~62 MB)
    float* ws     = (float*)d_ws;
    float* deg    = ws;
    float* agg    = deg + 40960;
    float* h1     = agg + (size_t)NNODES * NF;
    float* h2     = h1  + (size_t)NNODES * NF;
    float* pooled = h2  + (size_t)NNODES * NF;

    const int aggN       = NNODES * NF;                 // 5,120,000
    const int zeroAggGrd = (aggN + 255) / 256;
    const int scatterGrd = (NEDGES * 32) / 256;         // 1 wave per edge
    const int gemmGrd    = NNODES / 16;                 // 2500

    zero_f32<<<(NNODES + 255) / 256, 256, 0, stream>>>(deg, NNODES);
    zero_f32<<<(NGRAPHS * 3 * NF + 255) / 256, 256, 0, stream>>>(pooled, NGRAPHS * 3 * NF);
    degree_kernel<<<(NEDGES + 255) / 256, 256, 0, stream>>>(dst, deg);

    // Layer 1
    zero_f32<<<zeroAggGrd, 256, 0, stream>>>(agg, aggN);
    scatter_kernel<<<scatterGrd, 256, 0, stream>>>(x, src, dst, agg);
    sage_gemm<<<gemmGrd, 256, 0, stream>>>(agg, deg, x, W1l, W1r, b1, batch, h1, pooled, 0);

    // Layer 2
    zero_f32<<<zeroAggGrd, 256, 0, stream>>>(agg, aggN);
    scatter_kernel<<<scatterGrd, 256, 0, stream>>>(h1, src, dst, agg);
    sage_gemm<<<gemmGrd, 256, 0, stream>>>(agg, deg, h1, W2l, W2r, b2, batch, h2, pooled, NF);

    // Layer 3 (h3 never materialized; only pooled epilogue)
    zero_f32<<<zeroAggGrd, 256, 0, stream>>>(agg, aggN);
    scatter_kernel<<<scatterGrd, 256, 0, stream>>>(h2, src, dst, agg);
    sage_gemm<<<gemmGrd, 256, 0, stream>>>(agg, deg, h2, W3l, W3r, b3, batch, nullptr, pooled, 2 * NF);

    // Readout
    final_gemm<<<NGRAPHS / 16, 256, 0, stream>>>(pooled, Wlin, blin, out);
}